// GATraj_29755533427098
// MI455X (gfx1250) — compile-verified
//
#include <hip/hip_runtime.h>
#include <hip/hip_bf16.h>
#include <math.h>

#define NN 768
#define DD 64
#define EPSV 1e-5f

typedef __attribute__((ext_vector_type(2))) float v2f;
typedef __attribute__((ext_vector_type(8))) float v8f;

__device__ __forceinline__ float sigmoidf_(float x) { return 1.0f / (1.0f + __expf(-x)); }

// ---------------------------------------------------------------------------
// Kernel 1: LN closed-form moments of the relativeLayer (9 scalars).
// u(pair)  = c0*m0 + c1*m1 + mb
// s(pair)  = c0^2 V00 + c1^2 V11 + Vbb + 2(c0 c1 V01 + c0 V0b + c1 V1b)
// ---------------------------------------------------------------------------
__global__ __launch_bounds__(64) void k_stats(const float* __restrict__ W_r,
                                              const float* __restrict__ b_r,
                                              float* __restrict__ stats) {
    __shared__ float red[9][64];
    const int d = threadIdx.x;
    float w0 = W_r[d], w1 = W_r[64 + d], bb = b_r[d];
    red[0][d] = w0;      red[1][d] = w1;      red[2][d] = bb;
    red[3][d] = w0 * w0; red[4][d] = w1 * w1; red[5][d] = bb * bb;
    red[6][d] = w0 * w1; red[7][d] = w0 * bb; red[8][d] = w1 * bb;
    __syncthreads();
    for (int off = 32; off > 0; off >>= 1) {
        if (d < off)
            for (int q = 0; q < 9; ++q) red[q][d] += red[q][d + off];
        __syncthreads();
    }
    if (d == 0) {
        const float inv = 1.0f / 64.0f;
        float m0 = red[0][0] * inv, m1 = red[1][0] * inv, mb = red[2][0] * inv;
        stats[0] = m0; stats[1] = m1; stats[2] = mb;
        stats[3] = red[3][0] * inv - m0 * m0;   // V00
        stats[4] = red[4][0] * inv - m1 * m1;   // V11
        stats[5] = red[5][0] * inv - mb * mb;   // Vbb
        stats[6] = red[6][0] * inv - m0 * m1;   // V01
        stats[7] = red[7][0] * inv - m0 * mb;   // V0b
        stats[8] = red[8][0] * inv - m1 * mb;   // V1b
    }
}

// ---------------------------------------------------------------------------
// Kernel 2: A[i,d] = b_g[d] + h[i,:]@W_g[64:128,d]; B[j,d] = h[j,:]@W_g[128:192,d]
// ---------------------------------------------------------------------------
__global__ __launch_bounds__(256) void k_ab(const float* __restrict__ hidden,
                                            const float* __restrict__ W_g,
                                            const float* __restrict__ b_g,
                                            float* __restrict__ Ai,
                                            float* __restrict__ Bj) {
    int idx = blockIdx.x * 256 + threadIdx.x;
    if (idx >= NN * DD) return;
    int i = idx >> 6, d = idx & 63;
    float a = b_g[d], b = 0.f;
    for (int k = 0; k < 64; ++k) {
        float hv = hidden[i * 64 + k];
        a = fmaf(hv, W_g[(64 + k) * 64 + d], a);
        b = fmaf(hv, W_g[(128 + k) * 64 + d], b);
    }
    Ai[idx] = a;
    Bj[idx] = b;
}

// ---------------------------------------------------------------------------
// Kernel 3: collapsed softmax weight. tt == relu(be_a) (LN over size-1 axis).
// ---------------------------------------------------------------------------
__global__ __launch_bounds__(256) void k_invk(const int* __restrict__ nei,
                                              const float* __restrict__ be_a,
                                              float* __restrict__ invk) {
    int i = blockIdx.x * 256 + threadIdx.x;
    if (i >= NN) return;
    int cnt = 0;
    for (int j = 0; j < NN; ++j) cnt += (nei[i * NN + j] > 0) ? 1 : 0;
    float t = fmaxf(be_a[0], 0.f);
    float inv;
    if (t > 0.f)
        inv = (cnt > 0) ? (1.f / (float)cnt) : 0.f;   // uniform over neighbors
    else
        inv = 1.f / (float)NN;                        // all -10000 -> uniform row
    invk[i] = inv;
}

// ---------------------------------------------------------------------------
// Kernel 4: main per-pair kernel. One wave per i-row; loops 48 tiles of 16 j.
//   A(16x64) = r[i, j0..j0+15, :]   (built in registers, WMMA A f32 layout)
//   B(64x64) = W_g0                 (LDS, pair-interleaved for the B layout)
//   D = A@B via 16 K-steps x 4 N-tiles of V_WMMA_F32_16X16X4_F32
// then per-row LN+sigmoid gate, masked h[j]*gate accumulation, *1/k at end.
// ---------------------------------------------------------------------------
__global__ __launch_bounds__(128) void k_main(
    const float* __restrict__ corr, const int* __restrict__ nei,
    const float* __restrict__ hidden,
    const float* __restrict__ W_r, const float* __restrict__ b_r,
    const float* __restrict__ g_r, const float* __restrict__ be_r,
    const float* __restrict__ W_g,
    const float* __restrict__ g_g, const float* __restrict__ be_g,
    const float* __restrict__ stats, const float* __restrict__ Ai,
    const float* __restrict__ Bj, const float* __restrict__ invk,
    float* __restrict__ Hsum) {
    // B operand in WMMA layout: for step s, half h, col c the pair
    // (W_g0[4s+2h][c], W_g0[4s+2h+1][c]) is contiguous.
    __shared__ float2 sWB[2048];   // 16 KB
    __shared__ float4 sRQ[64];     // (W_r0, W_r1, b_r, g_r)
    __shared__ float  sBE[64];     // be_r

    const int tid = threadIdx.x;
    for (int idx = tid; idx < 4096; idx += 128) {
        int k = idx >> 6, col = idx & 63;
        int s = k >> 2, rem = k & 3, hh = rem >> 1, comp = rem & 1;
        ((float*)sWB)[(((s << 1) + hh) * 64 + col) * 2 + comp] = W_g[k * 64 + col];
    }
    for (int d = tid; d < 64; d += 128) {
        sRQ[d] = make_float4(W_r[d], W_r[64 + d], b_r[d], g_r[d]);
        sBE[d] = be_r[d];
    }
    __syncthreads();

    const int wave = tid >> 5, lane = tid & 31;
    const int i = blockIdx.x * 4 + wave;
    const int l16 = lane & 15, half = lane >> 4;

    const float m0 = stats[0], m1 = stats[1], mb = stats[2];
    const float V00 = stats[3], V11 = stats[4], Vbb = stats[5];
    const float V01 = stats[6], V0b = stats[7], V1b = stats[8];

    float aival[4], ggv[4], bev[4], Hacc[4];
#pragma unroll
    for (int n = 0; n < 4; ++n) {
        int dn = (n << 4) + l16;
        aival[n] = Ai[i * 64 + dn];
        ggv[n] = g_g[dn];
        bev[n] = be_g[dn];
        Hacc[n] = 0.f;
    }

    for (int jt = 0; jt < 48; ++jt) {
        const int j0 = jt << 4;
        const int jme = j0 + l16;                 // this lane's A-row pair (i, jme)
        const float c0 = corr[(i * NN + jme) * 2 + 0];
        const float c1 = corr[(i * NN + jme) * 2 + 1];
        const float u = fmaf(c0, m0, fmaf(c1, m1, mb));
        const float var = c0 * c0 * V00 + c1 * c1 * V11 + Vbb +
                          2.f * (c0 * c1 * V01 + c0 * V0b + c1 * V1b);
        const float rs = rsqrtf(var + EPSV);

        v8f acc0 = {}, acc1 = {}, acc2 = {}, acc3 = {};

#pragma unroll
        for (int s = 0; s < 16; ++s) {
            const int d0 = (s << 2) + (half << 1);           // K for this lane half
            float4 q0 = sRQ[d0], q1 = sRQ[d0 + 1];
            float v0 = fmaf(c0, q0.x, fmaf(c1, q0.y, q0.z));
            float v1 = fmaf(c0, q1.x, fmaf(c1, q1.y, q1.z));
            float rv0 = fmaxf(fmaf(q0.w * (v0 - u), rs, sBE[d0]), 0.f);
            float rv1 = fmaxf(fmaf(q1.w * (v1 - u), rs, sBE[d0 + 1]), 0.f);
            v2f a; a.x = rv0; a.y = rv1;

            const int base = ((s << 1) + half) * 64 + l16;
            float2 b0 = sWB[base + 0];
            float2 b1 = sWB[base + 16];
            float2 b2 = sWB[base + 32];
            float2 b3 = sWB[base + 48];
            v2f vb;
            vb.x = b0.x; vb.y = b0.y;
            acc0 = __builtin_amdgcn_wmma_f32_16x16x4_f32(false, a, false, vb, (short)0, acc0, false, false);
            vb.x = b1.x; vb.y = b1.y;
            acc1 = __builtin_amdgcn_wmma_f32_16x16x4_f32(false, a, false, vb, (short)0, acc1, false, false);
            vb.x = b2.x; vb.y = b2.y;
            acc2 = __builtin_amdgcn_wmma_f32_16x16x4_f32(false, a, false, vb, (short)0, acc2, false, false);
            vb.x = b3.x; vb.y = b3.y;
            acc3 = __builtin_amdgcn_wmma_f32_16x16x4_f32(false, a, false, vb, (short)0, acc3, false, false);
        }

        // Gate stage. C layout: lanes<16 hold rows p(0..7), lanes>=16 rows p+8.
#pragma unroll
        for (int p = 0; p < 8; ++p) {
            const int row = p + (half << 3);
            const int j = j0 + row;
            const bool msk = nei[i * NN + j] > 0;
            float x0 = acc0[p] + aival[0] + Bj[j * 64 + 0  + l16];
            float x1 = acc1[p] + aival[1] + Bj[j * 64 + 16 + l16];
            float x2 = acc2[p] + aival[2] + Bj[j * 64 + 32 + l16];
            float x3 = acc3[p] + aival[3] + Bj[j * 64 + 48 + l16];
            float sum = x0 + x1 + x2 + x3;
            float ss = x0 * x0 + x1 * x1 + x2 * x2 + x3 * x3;
#pragma unroll
            for (int m = 1; m < 16; m <<= 1) {    // stays within 16-lane group
                sum += __shfl_xor(sum, m, 32);
                ss  += __shfl_xor(ss,  m, 32);
            }
            const float u2 = sum * (1.f / 64.f);
            const float rs2 = rsqrtf(ss * (1.f / 64.f) - u2 * u2 + EPSV);
            if (msk) {
                float g0 = sigmoidf_(fmaf(ggv[0] * (x0 - u2), rs2, bev[0]));
                float g1 = sigmoidf_(fmaf(ggv[1] * (x1 - u2), rs2, bev[1]));
                float g2 = sigmoidf_(fmaf(ggv[2] * (x2 - u2), rs2, bev[2]));
                float g3 = sigmoidf_(fmaf(ggv[3] * (x3 - u2), rs2, bev[3]));
                Hacc[0] = fmaf(hidden[j * 64 + 0  + l16], g0, Hacc[0]);
                Hacc[1] = fmaf(hidden[j * 64 + 16 + l16], g1, Hacc[1]);
                Hacc[2] = fmaf(hidden[j * 64 + 32 + l16], g2, Hacc[2]);
                Hacc[3] = fmaf(hidden[j * 64 + 48 + l16], g3, Hacc[3]);
            }
        }
    }

    // combine lower-row (lanes<16) and upper-row (lanes>=16) partials per d
#pragma unroll
    for (int n = 0; n < 4; ++n) Hacc[n] += __shfl_xor(Hacc[n], 16, 32);
    const float inv = invk[i];
    if (lane < 16) {
#pragma unroll
        for (int n = 0; n < 4; ++n) Hsum[i * 64 + (n << 4) + lane] = Hacc[n] * inv;
    }
}

// ---------------------------------------------------------------------------
// Kernel 5: C = relu(LN(Hsum@W_w + b_w)) + cn ; Hout = h + tanh(C)
// One wave per row; lane owns d and d+32.
// ---------------------------------------------------------------------------
__global__ __launch_bounds__(256) void k_final(
    const float* __restrict__ Hsum, const float* __restrict__ hidden,
    const float* __restrict__ cn,
    const float* __restrict__ W_w, const float* __restrict__ b_w,
    const float* __restrict__ g_w, const float* __restrict__ be_w,
    float* __restrict__ Hout, float* __restrict__ Cout) {
    const int wave = threadIdx.x >> 5, lane = threadIdx.x & 31;
    const int row = blockIdx.x * 8 + wave;
    if (row >= NN) return;
    const int d0 = lane, d1 = lane + 32;
    float a0 = b_w[d0], a1 = b_w[d1];
    for (int k = 0; k < 64; ++k) {
        float hs = Hsum[row * 64 + k];
        a0 = fmaf(hs, W_w[k * 64 + d0], a0);
        a1 = fmaf(hs, W_w[k * 64 + d1], a1);
    }
    float sum = a0 + a1, ss = a0 * a0 + a1 * a1;
#pragma unroll
    for (int m = 1; m < 32; m <<= 1) {
        sum += __shfl_xor(sum, m, 32);
        ss  += __shfl_xor(ss,  m, 32);
    }
    float u = sum * (1.f / 64.f);
    float rs = rsqrtf(ss * (1.f / 64.f) - u * u + EPSV);
    float c0 = fmaxf(fmaf(g_w[d0] * (a0 - u), rs, be_w[d0]), 0.f) + cn[row * 64 + d0];
    float c1 = fmaxf(fmaf(g_w[d1] * (a1 - u), rs, be_w[d1]), 0.f) + cn[row * 64 + d1];
    Cout[row * 64 + d0] = c0;
    Cout[row * 64 + d1] = c1;
    Hout[row * 64 + d0] = hidden[row * 64 + d0] + tanhf(c0);
    Hout[row * 64 + d1] = hidden[row * 64 + d1] + tanhf(c1);
}

// ---------------------------------------------------------------------------
extern "C" void kernel_launch(void* const* d_in, const int* in_sizes, int n_in,
                              void* d_out, int out_size, void* d_ws, size_t ws_size,
                              hipStream_t stream) {
    const float* corr   = (const float*)d_in[0];
    const int*   nei    = (const int*)d_in[1];
    // d_in[2] = nei_num : unused by the reference computation
    const float* hidden = (const float*)d_in[3];
    const float* cn     = (const float*)d_in[4];
    const float* W_r  = (const float*)d_in[5];
    const float* b_r  = (const float*)d_in[6];
    const float* g_r  = (const float*)d_in[7];
    const float* be_r = (const float*)d_in[8];
    const float* W_g  = (const float*)d_in[9];
    const float* b_g  = (const float*)d_in[10];
    const float* g_g  = (const float*)d_in[11];
    const float* be_g = (const float*)d_in[12];
    // d_in[13..15] = W_a, b_a, g_a : provably unused (LN over size-1 axis)
    const float* be_a = (const float*)d_in[16];
    const float* W_w  = (const float*)d_in[17];
    const float* b_w  = (const float*)d_in[18];
    const float* g_w  = (const float*)d_in[19];
    const float* be_w = (const float*)d_in[20];

    float* ws    = (float*)d_ws;
    float* stats = ws;                 // 16 floats (9 used)
    float* Ai    = ws + 16;            // 768*64
    float* Bj    = Ai + NN * DD;       // 768*64
    float* Hs    = Bj + NN * DD;       // 768*64
    float* invk  = Hs + NN * DD;       // 768

    float* Hout = (float*)d_out;       // output 0: H_out [768,64]
    float* Cout = Hout + NN * DD;      // output 1: C     [768,64]

    k_stats<<<1, 64, 0, stream>>>(W_r, b_r, stats);
    k_ab<<<(NN * DD + 255) / 256, 256, 0, stream>>>(hidden, W_g, b_g, Ai, Bj);
    k_invk<<<(NN + 255) / 256, 256, 0, stream>>>(nei, be_a, invk);
    k_main<<<NN / 4, 128, 0, stream>>>(corr, nei, hidden, W_r, b_r, g_r, be_r,
                                       W_g, g_g, be_g, stats, Ai, Bj, invk, Hs);
    k_final<<<NN / 8, 256, 0, stream>>>(Hs, hidden, cn, W_w, b_w, g_w, be_w,
                                        Hout, Cout);
}